// ValidatedSpikeBridge_19748259627071
// MI455X (gfx1250) — compile-verified
//
#include <hip/hip_runtime.h>
#include <stdint.h>
#include <math.h>

// Temporal-contrast spike encoding for MI455X (gfx1250).
// Bandwidth-bound: ~32 MiB reads + 256 MiB NT writes => ~13 us roofline @ 23.3 TB/s.
// No matmul FLOPs -> no WMMA; CDNA5 paths used: async global->LDS (ASYNCcnt),
// s_wait_asynccnt, global_prefetch_b8, non-temporal b128 stores.

typedef float v4f __attribute__((ext_vector_type(4)));

#define BB 8
#define SS 4096
#define FF 128
#define TT 16
#define NCHUNK 64           // S-chunks per batch for the reduction pass
#define CH (SS / NCHUNK)    // 64 rows per chunk
#define THRESH 0.1f
#define EPSV 1e-8f

// ws layout (floats):
//   [0, BB*NCHUNK*FF)                      partial sums of diff
//   [BB*NCHUNK*FF, 2*BB*NCHUNK*FF)        partial sums of diff^2
//   [2*BB*NCHUNK*FF, +BB*FF)              per-(b,f) threshold = 0.1*(std+eps)
#define PARTN (BB * NCHUNK * FF)

// ---------------------------------------------------------------------------
// Pass 1: deterministic partial reduction of sum(d), sum(d^2) over S-chunks.
// 128 threads = one f-channel each; fully coalesced 512B row loads.
// ---------------------------------------------------------------------------
__global__ void tc_partial(const float* __restrict__ x, float* __restrict__ ws) {
  const int f = threadIdx.x;          // 0..127
  const int c = blockIdx.x;           // chunk id 0..63
  const int b = blockIdx.y;           // batch
  const float* base = x + ((size_t)b * SS) * FF + f;

  const int s0 = c * CH;
  const int send = s0 + CH;
  float sum = 0.0f, sq = 0.0f;
  float prev;
  int s = s0;
  if (s0 == 0) {              // diff[:,0,:] == 0 (prepend semantics)
    prev = base[0];
    s = 1;
  } else {
    prev = base[(size_t)(s0 - 1) * FF];
  }

  // Stream-ahead prefetch of the next chunk (lowers to global_prefetch_b8).
  if (send < SS) __builtin_prefetch(base + (size_t)send * FF, 0, 0);

  for (; s < send; ++s) {
    float cur = base[(size_t)s * FF];
    float d = cur - prev;
    prev = cur;
    sum += d;
    sq = fmaf(d, d, sq);
  }

  const int part = (b * NCHUNK + c) * FF + f;
  ws[part] = sum;
  ws[PARTN + part] = sq;
}

// ---------------------------------------------------------------------------
// Pass 2: fold the 64 chunk-partials per (b,f) in a fixed order (deterministic)
// and produce thr = THRESH * (std + eps). Population std via E[d^2]-E[d]^2.
// ---------------------------------------------------------------------------
__global__ void tc_thresh(const float* __restrict__ ws, float* __restrict__ thr) {
  const int bf = blockIdx.x * blockDim.x + threadIdx.x;
  if (bf >= BB * FF) return;
  const int b = bf / FF;
  const int f = bf % FF;
  float sum = 0.0f, sq = 0.0f;
  for (int c = 0; c < NCHUNK; ++c) {
    const int part = (b * NCHUNK + c) * FF + f;
    sum += ws[part];
    sq += ws[PARTN + part];
  }
  const float invS = 1.0f / (float)SS;
  const float mean = sum * invS;
  const float var = fmaxf(sq * invS - mean * mean, 0.0f);
  thr[bf] = THRESH * (sqrtf(var) + EPSV);
}

// ---------------------------------------------------------------------------
// Pass 3: recompute diff, threshold, expand to [B,T,S,F] with t%3 phase.
// Per-block: async-DMA the 512B threshold row for this batch into LDS
// (global_load_async_to_lds_b128, ASYNCcnt), then each thread handles one
// float4 of channels and emits 16 coalesced non-temporal b128 plane stores.
// ---------------------------------------------------------------------------
__global__ void tc_expand(const float* __restrict__ x,
                          const float* __restrict__ thr,
                          float* __restrict__ out) {
  __shared__ alignas(16) float s_thr[FF];
  const int b = blockIdx.y;

  // Wave 0 (lanes 0..31, wave32) pulls thr[b, 0:128] into LDS: 32 lanes x 16B.
  if (threadIdx.x < 32) {
    uint64_t g = (uint64_t)(uintptr_t)(thr + (size_t)b * FF) +
                 (uint64_t)threadIdx.x * 16u;
    uint32_t l = (uint32_t)(uintptr_t)(&s_thr[0]) + (uint32_t)threadIdx.x * 16u;
    asm volatile("global_load_async_to_lds_b128 %0, %1, off"
                 :: "v"(l), "v"(g)
                 : "memory");
    asm volatile("s_wait_asynccnt 0x0" ::: "memory");
  }
  __syncthreads();

  const int srow = threadIdx.x >> 5;        // 0..7   (8 S-rows per block)
  const int f4 = threadIdx.x & 31;          // 0..31  (float4 channel group)
  const int s = blockIdx.x * 8 + srow;
  const int f = f4 * 4;

  const float* px = x + (((size_t)b * SS) + s) * FF + f;
  v4f cur = *(const v4f*)px;
  v4f prv = (s > 0) ? *(const v4f*)(px - FF) : cur;  // s==0 -> d=0 -> no spike
  v4f d = cur - prv;
  v4f th = *(const v4f*)(&s_thr[f]);

  v4f pos, neg;
  pos.x = (d.x >= th.x) ? 1.0f : 0.0f;
  pos.y = (d.y >= th.y) ? 1.0f : 0.0f;
  pos.z = (d.z >= th.z) ? 1.0f : 0.0f;
  pos.w = (d.w >= th.w) ? 1.0f : 0.0f;
  neg.x = (d.x <= -th.x) ? 1.0f : 0.0f;
  neg.y = (d.y <= -th.y) ? 1.0f : 0.0f;
  neg.z = (d.z <= -th.z) ? 1.0f : 0.0f;
  neg.w = (d.w <= -th.w) ? 1.0f : 0.0f;
  v4f zero = {0.0f, 0.0f, 0.0f, 0.0f};

  float* po = out + ((((size_t)b * TT) * SS) + s) * FF + f;
  const size_t plane = (size_t)SS * FF;
#pragma unroll
  for (int t = 0; t < TT; ++t) {
    const int ph = t % 3;
    v4f v = (ph == 0) ? pos : (ph == 1) ? neg : zero;
    // Streaming 256 MiB output: non-temporal b128 stores keep L2 clean.
    __builtin_nontemporal_store(v, (v4f*)(po + (size_t)t * plane));
  }
}

// ---------------------------------------------------------------------------
extern "C" void kernel_launch(void* const* d_in, const int* in_sizes, int n_in,
                              void* d_out, int out_size, void* d_ws, size_t ws_size,
                              hipStream_t stream) {
  (void)in_sizes; (void)n_in; (void)out_size; (void)ws_size;
  const float* x = (const float*)d_in[0];   // [8, 4096, 128] fp32
  float* out = (float*)d_out;               // [8, 16, 4096, 128] fp32
  float* ws = (float*)d_ws;                 // needs ~516 KB
  float* thr = ws + 2 * PARTN;

  // Pass 1: partial reductions (no atomics -> deterministic).
  tc_partial<<<dim3(NCHUNK, BB), FF, 0, stream>>>(x, ws);
  // Pass 2: per-(b,f) threshold.
  tc_thresh<<<dim3((BB * FF + 255) / 256), 256, 0, stream>>>(ws, thr);
  // Pass 3: encode + temporal expand, NT stores.
  tc_expand<<<dim3(SS / 8, BB), 256, 0, stream>>>(x, thr, out);
}